// DESimplE_18786186953557
// MI455X (gfx1250) — compile-verified
//
#include <hip/hip_runtime.h>

#define S_DIM 384
#define T_DIM 128
#define F_DIM 512   // S_DIM + T_DIM

struct Params {
    const int   *s, *r, *o;
    const float *ty, *tm, *td;
    const float *es, *eo, *rf, *ri;
    const float *tp[18];   // index = p*6 + kind*2 + sfx ; p:{y,m,d} kind:{frq,phi,amp} sfx:{s,o}
    float       *out;
    int          n;
};

__device__ __forceinline__ float4 ld4(const float* p) {
    return *reinterpret_cast<const float4*>(p);
}

// acc += sum_i a_i * r_i * b_i  (per float4)
__device__ __forceinline__ void fma3(float& acc, const float4 a, const float4 r, const float4 b) {
    acc = fmaf(a.x * r.x, b.x, acc);
    acc = fmaf(a.y * r.y, b.y, acc);
    acc = fmaf(a.z * r.z, b.z, acc);
    acc = fmaf(a.w * r.w, b.w, acc);
}

// 4 lanes' worth of the temporal embedding: sum_p amp*sin(frq*t_p + phi)
__device__ __forceinline__ float4 temb(const Params& P, int sfx, size_t e,
                                       float t0, float t1, float t2, int off) {
    float4 acc = make_float4(0.f, 0.f, 0.f, 0.f);
    const float t[3] = {t0, t1, t2};
    const size_t base = e * (size_t)T_DIM + (size_t)off;
#pragma unroll
    for (int p = 0; p < 3; ++p) {
        const float4 fr = ld4(P.tp[p * 6 + 0 + sfx] + base);  // frq
        const float4 ph = ld4(P.tp[p * 6 + 2 + sfx] + base);  // phi
        const float4 am = ld4(P.tp[p * 6 + 4 + sfx] + base);  // amp
        acc.x = fmaf(am.x, __sinf(fmaf(fr.x, t[p], ph.x)), acc.x);
        acc.y = fmaf(am.y, __sinf(fmaf(fr.y, t[p], ph.y)), acc.y);
        acc.z = fmaf(am.z, __sinf(fmaf(fr.z, t[p], ph.z)), acc.z);
        acc.w = fmaf(am.w, __sinf(fmaf(fr.w, t[p], ph.w)), acc.w);
    }
    return acc;
}

__global__ void __launch_bounds__(256)
desimple_score_kernel(Params P) {
    // One wave32 per item. item is uniform across the wave by construction;
    // readfirstlane proves it to the compiler so the 6 per-item scalars become
    // s_load_b32 (SMEM/KMcnt path) and all row bases stay in SGPRs.
    const int item = __builtin_amdgcn_readfirstlane(
        (int)((blockIdx.x * blockDim.x + threadIdx.x) >> 5));
    const int lane = (int)(threadIdx.x & 31);
    if (item >= P.n) return;

    const size_t si = (size_t)P.s[item];
    const size_t rr = (size_t)P.r[item];
    const size_t oi = (size_t)P.o[item];
    const float  t0 = P.ty[item], t1 = P.tm[item], t2 = P.td[item];

    const int toff = lane * 4;  // float4 per lane covers 128 dims per pass

    // --- CDNA5 prefetch (global_prefetch_b8): launch the 36 temporal rows and the
    // relation tails now; they land in cache while the entity part executes below.
#pragma unroll
    for (int p = 0; p < 3; ++p) {
#pragma unroll
        for (int k = 0; k < 3; ++k) {
            __builtin_prefetch(P.tp[p * 6 + k * 2 + 0] + si * T_DIM + toff, 0, 3); // sfx s @ s
            __builtin_prefetch(P.tp[p * 6 + k * 2 + 1] + oi * T_DIM + toff, 0, 3); // sfx o @ o
            __builtin_prefetch(P.tp[p * 6 + k * 2 + 0] + oi * T_DIM + toff, 0, 3); // sfx s @ o
            __builtin_prefetch(P.tp[p * 6 + k * 2 + 1] + si * T_DIM + toff, 0, 3); // sfx o @ s
        }
    }
    __builtin_prefetch(P.rf + rr * F_DIM + S_DIM + toff, 0, 3);
    __builtin_prefetch(P.ri + rr * F_DIM + S_DIM + toff, 0, 3);

    float acc = 0.f;

    // --- entity part: 384 dims = 3 coalesced float4 passes per lane
    const float* a0 = P.es + si * S_DIM;   // s_emb_s (struct part)
    const float* b0 = P.eo + oi * S_DIM;   // o_emb_o
    const float* c0 = P.es + oi * S_DIM;   // o_emb_s
    const float* d0 = P.eo + si * S_DIM;   // s_emb_o
    const float* f0 = P.rf + rr * F_DIM;
    const float* g0 = P.ri + rr * F_DIM;
#pragma unroll
    for (int pass = 0; pass < 3; ++pass) {
        const int off = pass * 128 + toff;
        const float4 a = ld4(a0 + off);
        const float4 b = ld4(b0 + off);
        const float4 c = ld4(c0 + off);
        const float4 d = ld4(d0 + off);
        const float4 f = ld4(f0 + off);
        const float4 g = ld4(g0 + off);
        fma3(acc, a, f, b);
        fma3(acc, c, g, d);
    }

    // --- temporal part: 128 dims = 1 float4 pass per lane
    {
        const float4 f = ld4(f0 + S_DIM + toff);
        const float4 g = ld4(g0 + S_DIM + toff);
        const float4 A = temb(P, 0, si, t0, t1, t2, toff);  // t_emb_s(s)
        const float4 B = temb(P, 1, oi, t0, t1, t2, toff);  // t_emb_o(o)
        const float4 C = temb(P, 0, oi, t0, t1, t2, toff);  // t_emb_s(o)
        const float4 D = temb(P, 1, si, t0, t1, t2, toff);  // t_emb_o(s)
        fma3(acc, A, f, B);
        fma3(acc, C, g, D);
    }

    // --- wave32 reduction
#pragma unroll
    for (int offs = 16; offs > 0; offs >>= 1)
        acc += __shfl_down(acc, offs, 32);

    if (lane == 0) P.out[item] = 0.5f * acc;
}

extern "C" void kernel_launch(void* const* d_in, const int* in_sizes, int n_in,
                              void* d_out, int out_size, void* d_ws, size_t ws_size,
                              hipStream_t stream) {
    (void)n_in; (void)out_size; (void)d_ws; (void)ws_size;
    Params P;
    P.s  = (const int*)  d_in[0];
    P.r  = (const int*)  d_in[1];
    P.o  = (const int*)  d_in[2];
    P.ty = (const float*)d_in[3];
    P.tm = (const float*)d_in[4];
    P.td = (const float*)d_in[5];
    P.es = (const float*)d_in[6];
    P.eo = (const float*)d_in[7];
    P.rf = (const float*)d_in[8];
    P.ri = (const float*)d_in[9];
    for (int i = 0; i < 18; ++i) P.tp[i] = (const float*)d_in[10 + i];
    P.out = (float*)d_out;
    P.n   = in_sizes[0];

    const int items_per_block = 8;               // 8 wave32 per 256-thread block
    const int grid = (P.n + items_per_block - 1) / items_per_block;
    hipLaunchKernelGGL(desimple_score_kernel, dim3(grid), dim3(256), 0, stream, P);
}